// PositionalEncodingOffGrid_3719441678422
// MI455X (gfx1250) — compile-verified
//
#include <hip/hip_runtime.h>

// 4-wide float vector with addressable-free component access + NT store support
typedef __attribute__((ext_vector_type(4))) float v4f;

// out[c*num + i], c in [0,256):
//   c = 2f        -> sin(y_i * w_f)
//   c = 2f+1      -> cos(y_i * w_f)
//   c = 2H+2f     -> sin(x_i * w_f)
//   c = 2H+2f+1   -> cos(x_i * w_f)     (H = half = |inv_freq| = 64)
__global__ __launch_bounds__(256) void pe_offgrid_kernel(
    const float* __restrict__ YX,        // (2, num)
    const float* __restrict__ inv_freq,  // (half,)
    float* __restrict__ out,             // (4*half, num)
    int num, int half) {
  const long long i4 = 4LL * (blockIdx.x * (long long)blockDim.x + threadIdx.x);
  if (i4 >= num) return;

  const float* __restrict__ Y = YX;
  const float* __restrict__ X = YX + num;
  const size_t row = (size_t)num;

  if (i4 + 3 < num) {
    // ---- fast path: 4 positions per thread, b128 NT stores ----
    const v4f yv = *(const v4f*)(Y + i4);
    const v4f xv = *(const v4f*)(X + i4);
    const float ya[4] = {yv.x, yv.y, yv.z, yv.w};
    const float xa[4] = {xv.x, xv.y, xv.z, xv.w};

#pragma unroll 4
    for (int f = 0; f < half; ++f) {
      const float w = inv_freq[f];  // uniform -> s_load through K$
      float ys[4], yc[4], xs[4], xc[4];
#pragma unroll
      for (int j = 0; j < 4; ++j) {
        __sincosf(ya[j] * w, &ys[j], &yc[j]);
        __sincosf(xa[j] * w, &xs[j], &xc[j]);
      }
      const v4f vsy = {ys[0], ys[1], ys[2], ys[3]};
      const v4f vcy = {yc[0], yc[1], yc[2], yc[3]};
      const v4f vsx = {xs[0], xs[1], xs[2], xs[3]};
      const v4f vcx = {xc[0], xc[1], xc[2], xc[3]};

      float* py = out + (size_t)(2 * f) * row + i4;
      __builtin_nontemporal_store(vsy, (v4f*)py);
      __builtin_nontemporal_store(vcy, (v4f*)(py + row));
      float* px = out + (size_t)(2 * half + 2 * f) * row + i4;
      __builtin_nontemporal_store(vsx, (v4f*)px);
      __builtin_nontemporal_store(vcx, (v4f*)(px + row));
    }
  } else {
    // ---- tail path (num % 4 != 0): scalar, bounds-checked ----
    for (int j = 0; j < 4 && (i4 + j) < num; ++j) {
      const float y = Y[i4 + j];
      const float x = X[i4 + j];
      for (int f = 0; f < half; ++f) {
        const float w = inv_freq[f];
        float sy, cy, sx, cx;
        __sincosf(y * w, &sy, &cy);
        __sincosf(x * w, &sx, &cx);
        out[(size_t)(2 * f) * row + i4 + j] = sy;
        out[(size_t)(2 * f + 1) * row + i4 + j] = cy;
        out[(size_t)(2 * half + 2 * f) * row + i4 + j] = sx;
        out[(size_t)(2 * half + 2 * f + 1) * row + i4 + j] = cx;
      }
    }
  }
}

extern "C" void kernel_launch(void* const* d_in, const int* in_sizes, int n_in,
                              void* d_out, int out_size, void* d_ws, size_t ws_size,
                              hipStream_t stream) {
  const float* YX = (const float*)d_in[0];        // (2, num) float32
  const float* inv_freq = (const float*)d_in[1];  // (half,)  float32
  float* out = (float*)d_out;                     // (4*half, num) float32

  const int num = in_sizes[0] / 2;   // 262144
  const int half = in_sizes[1];      // 64

  const int threads_needed = (num + 3) / 4;
  const int block = 256;             // 8 wave32 waves per workgroup
  const int grid = (threads_needed + block - 1) / block;

  pe_offgrid_kernel<<<grid, block, 0, stream>>>(YX, inv_freq, out, num, half);
}